// LSTM_22488448762672
// MI455X (gfx1250) — compile-verified
//
#include <hip/hip_runtime.h>
#include <hip/hip_bf16.h>
#include <math.h>

// ---------------------------------------------------------------------------
// 2-layer LSTM, T=256, B=32, H=1024, VOCAB=8192, fp32.
// Strategy (MI455X):
//  * All weights (~182MB) fit in the 192MB L2 -> per-step weight re-reads are
//    L2 hits; fuse GEMM + bias + embedding gather + LSTM cell per time step.
//  * fp32 WMMA (v_wmma_f32_16x16x4_f32) to match reference numerics exactly
//    (bf16 would drift over 256 recurrent steps).
//  * Per step-kernel: 64 blocks x 128 threads. Block owns 16 hidden columns;
//    wave g (0..3) computes gate g's 32x16 tile over K=1024 (256 wmma blocks,
//    2 wmma per block for batch rows 0-15 / 16-31). 8KB LDS gate exchange,
//    then the LSTM cell, h/c update, and y1 store — all in one kernel.
//  * Layer 1 fuses its input GEMM (y0[t] @ W_ih1^T) as a second K loop.
//  * Workspace: only 6 ping-pong state buffers (768 KB).
// ---------------------------------------------------------------------------

#define T_STEPS 256
#define BATCH   32
#define HID     1024
#define VOCAB   8192

typedef __attribute__((ext_vector_type(2))) float v2f;
typedef __attribute__((ext_vector_type(8))) float v8f;

__device__ __forceinline__ v8f wmma_f32_k4(v2f a, v2f b, v8f c) {
  // D(16x16,f32) = A(16x4,f32) * B(4x16,f32) + C
  return __builtin_amdgcn_wmma_f32_16x16x4_f32(
      /*neg_a=*/false, a, /*neg_b=*/false, b,
      /*c_mod=*/(short)0, c, /*reuse_a=*/false, /*reuse_b=*/false);
}

__device__ __forceinline__ float sigmoidf_(float x) {
  return 1.0f / (1.0f + expf(-x));
}

// One LSTM time step for one layer, fused.
//   gates(32,4096) = h_in @ W_hh^T  [+ x_in @ W_ih^T (layer1)]
//                    [+ gather W_ih0[:, tok[b]] (layer0)] + b_ih + b_hh
//   then cell nonlinearity, c updated in place, h_out (+ optional y_out).
template <int LAYER>
__global__ __launch_bounds__(128) void lstm_step_kernel(
    const float* __restrict__ h_in,    // (32, 1024)
    const float* __restrict__ x_in,    // layer1: (32, 1024) = y0[t]; else null
    const int*   __restrict__ tok,     // layer0: 32 token ids; else null
    const float* __restrict__ W_hh,    // (4096, 1024) row-major
    const float* __restrict__ W_ih,    // layer0: (4096, 8192); layer1: (4096,1024)
    const float* __restrict__ b_ih,    // (4096)
    const float* __restrict__ b_hh,    // (4096)
    float*       __restrict__ c_state, // (32, 1024) in-place
    float*       __restrict__ h_out,   // (32, 1024)
    float*       __restrict__ y_out)   // (32, 1024) or null
{
  const int tid   = threadIdx.x;
  const int lane  = tid & 31;
  const int gate  = tid >> 5;        // wave id = gate index (i,f,g,o)
  const int hi    = lane >> 4;       // lane half selects K pair / row half
  const int jcol  = lane & 15;       // N within tile (also A-row within tile)
  const int jbase = blockIdx.x * 16; // hidden-column tile base
  const int n     = gate * HID + jbase + jcol; // gate column in [0,4096)
  const int kk    = 2 * hi;          // K offset inside each K=4 block

  v8f acc0 = {0.f, 0.f, 0.f, 0.f, 0.f, 0.f, 0.f, 0.f}; // batch rows 0..15
  v8f acc1 = {0.f, 0.f, 0.f, 0.f, 0.f, 0.f, 0.f, 0.f}; // batch rows 16..31

  // ---- recurrent GEMM: h_in @ W_hh^T -----------------------------------
  {
    const float* wrow  = W_hh + (size_t)n * HID;
    const float* arow0 = h_in + (size_t)jcol * HID;
    const float* arow1 = h_in + (size_t)(jcol + 16) * HID;
#pragma unroll 4
    for (int k0 = 0; k0 < HID; k0 += 4) {
      v2f bf = *(const v2f*)(wrow  + k0 + kk);
      v2f a0 = *(const v2f*)(arow0 + k0 + kk);
      v2f a1 = *(const v2f*)(arow1 + k0 + kk);
      acc0 = wmma_f32_k4(a0, bf, acc0);
      acc1 = wmma_f32_k4(a1, bf, acc1);
    }
  }

  // ---- layer1: fused input GEMM: x_in @ W_ih^T -------------------------
  if (LAYER == 1) {
    const float* wrow  = W_ih + (size_t)n * HID;
    const float* arow0 = x_in + (size_t)jcol * HID;
    const float* arow1 = x_in + (size_t)(jcol + 16) * HID;
#pragma unroll 4
    for (int k0 = 0; k0 < HID; k0 += 4) {
      v2f bf = *(const v2f*)(wrow  + k0 + kk);
      v2f a0 = *(const v2f*)(arow0 + k0 + kk);
      v2f a1 = *(const v2f*)(arow1 + k0 + kk);
      acc0 = wmma_f32_k4(a0, bf, acc0);
      acc1 = wmma_f32_k4(a1, bf, acc1);
    }
  }

  // ---- epilogue: bias (+ one-hot gather for layer0) --------------------
  const float bias = b_ih[n] + b_hh[n];
  if (LAYER == 0) {
    // one_hot(x) @ W_ih0^T == gather column tok[b] of W_ih0 (L2-resident)
#pragma unroll
    for (int r = 0; r < 8; ++r) {
      const int b0 = r + 8 * hi;       // C/D layout: vgpr r, lane-half -> +8
      const int b1 = b0 + 16;
      acc0[r] += bias + W_ih[(size_t)n * VOCAB + tok[b0]];
      acc1[r] += bias + W_ih[(size_t)n * VOCAB + tok[b1]];
    }
  } else {
#pragma unroll
    for (int r = 0; r < 8; ++r) {
      acc0[r] += bias;
      acc1[r] += bias;
    }
  }

  // ---- exchange 4 gate tiles through LDS -------------------------------
  __shared__ float gsm[4][32][17]; // [gate][batch][jcol], padded
#pragma unroll
  for (int r = 0; r < 8; ++r) {
    gsm[gate][r + 8 * hi][jcol]      = acc0[r];
    gsm[gate][16 + r + 8 * hi][jcol] = acc1[r];
  }
  __syncthreads();

  // ---- LSTM cell: 32 batch x 16 cols = 512 elems, 4 per thread ---------
#pragma unroll
  for (int q = 0; q < 4; ++q) {
    const int e = tid + 128 * q;
    const int b = e >> 4;
    const int j = e & 15;
    const size_t idx = (size_t)b * HID + (jbase + j);
    const float ig = sigmoidf_(gsm[0][b][j]);
    const float fg = sigmoidf_(gsm[1][b][j]);
    const float gg = tanhf(gsm[2][b][j]);
    const float og = sigmoidf_(gsm[3][b][j]);
    const float cc = fg * c_state[idx] + ig * gg;
    c_state[idx] = cc;
    const float hh = og * tanhf(cc);
    h_out[idx] = hh;
    if (y_out) y_out[idx] = hh;
  }
}

__global__ __launch_bounds__(256) void copy_states_kernel(
    const float* __restrict__ h0, const float* __restrict__ h1,
    const float* __restrict__ c0, const float* __restrict__ c1,
    float* __restrict__ out)
{
  const int i = blockIdx.x * blockDim.x + threadIdx.x;
  const int NBH = BATCH * HID;                       // 32768
  const size_t base = (size_t)T_STEPS * BATCH * HID; // 8388608 (y1)
  if (i < NBH) {
    out[base + i]           = h0[i];
    out[base + NBH + i]     = h1[i];
    out[base + 2 * NBH + i] = c0[i];
    out[base + 3 * NBH + i] = c1[i];
  }
}

extern "C" void kernel_launch(void* const* d_in, const int* in_sizes, int n_in,
                              void* d_out, int out_size, void* d_ws, size_t ws_size,
                              hipStream_t stream) {
  const int*   x     = (const int*)  d_in[0];
  const float* Hst   = (const float*)d_in[1];  // (2, 32, 1024)
  const float* Cst   = (const float*)d_in[2];  // (2, 32, 1024)
  const float* W_ih0 = (const float*)d_in[3];  // (4096, 8192)
  const float* W_hh0 = (const float*)d_in[4];  // (4096, 1024)
  const float* b_ih0 = (const float*)d_in[5];
  const float* b_hh0 = (const float*)d_in[6];
  const float* W_ih1 = (const float*)d_in[7];  // (4096, 1024)
  const float* W_hh1 = (const float*)d_in[8];  // (4096, 1024)
  const float* b_ih1 = (const float*)d_in[9];
  const float* b_hh1 = (const float*)d_in[10];
  float* out = (float*)d_out;

  const int NBH = BATCH * HID; // 32768 floats
  float* ws = (float*)d_ws;
  float* h0buf[2] = { ws,            ws + NBH     };
  float* h1buf[2] = { ws + 2 * NBH,  ws + 3 * NBH };
  float* c0 = ws + 4 * NBH;
  float* c1 = ws + 5 * NBH;

  // Initial states (D2D async copies are graph-capture safe).
  hipMemcpyAsync(h0buf[0], Hst,       NBH * sizeof(float), hipMemcpyDeviceToDevice, stream);
  hipMemcpyAsync(h1buf[0], Hst + NBH, NBH * sizeof(float), hipMemcpyDeviceToDevice, stream);
  hipMemcpyAsync(c0,       Cst,       NBH * sizeof(float), hipMemcpyDeviceToDevice, stream);
  hipMemcpyAsync(c1,       Cst + NBH, NBH * sizeof(float), hipMemcpyDeviceToDevice, stream);

  for (int t = 0; t < T_STEPS; ++t) {
    const int pi = t & 1, po = (t + 1) & 1;
    lstm_step_kernel<0><<<HID / 16, 128, 0, stream>>>(
        h0buf[pi], nullptr, x + t * BATCH, W_hh0, W_ih0, b_ih0, b_hh0,
        c0, h0buf[po], nullptr);
    lstm_step_kernel<1><<<HID / 16, 128, 0, stream>>>(
        h1buf[pi], h0buf[po], nullptr, W_hh1, W_ih1, b_ih1, b_hh1,
        c1, h1buf[po], out + (size_t)t * NBH);
  }
  // After 256 steps final h lives in buffer 0 for both layers.
  copy_states_kernel<<<(NBH + 255) / 256, 256, 0, stream>>>(
      h0buf[0], h1buf[0], c0, c1, out);
}